// EKF_Lorenz_38019050504665
// MI455X (gfx1250) — compile-verified
//
#include <hip/hip_runtime.h>
#include <cstdint>

// ---------------- problem constants (from reference setup) ----------------
#define N_SEQ   4096
#define T_STEPS 1000
#define BLK     128                   // 4 wave32 per block
#define GRID_B  (N_SEQ / BLK)         // 32 blocks
#define CHUNK   8                     // timesteps per TDM tile
#define NCHUNK  (T_STEPS / CHUNK)     // 125, exact
#define ROWF    (CHUNK * 3)           // 24 floats of Y per lane per chunk (96 B rows)
#define PROWF   (CHUNK * 9)           // 72 floats of P per lane per chunk (288 B rows)
#define YBUF    (BLK * ROWF)          // 3072 floats per Y buffer (12 KB)
#define PBUF    (BLK * PROWF)         // 9216 floats per P buffer (36 KB)
#define PSTG_OFF (2 * YBUF)           // P staging starts after the two Y buffers
#define SMEM_FLOATS (2 * YBUF + 2 * PBUF)  // 24576 floats = 96 KB

typedef unsigned int v4u __attribute__((ext_vector_type(4)));
typedef int          v8i __attribute__((ext_vector_type(8)));
typedef int          v4i __attribute__((ext_vector_type(4)));

// ---------------- TDM descriptor: 2D tile, 128 rows, contiguous row of tile0 ----
__device__ __forceinline__ void tdm_desc(v4u& g0, v8i& g1, unsigned lds_byte,
                                         unsigned long long gaddr,
                                         unsigned dim0, unsigned stride0,
                                         unsigned tile0) {
  g0.x = 1u;                                              // count=1 (valid descriptor)
  g0.y = lds_byte;                                        // LDS byte address
  g0.z = (unsigned)(gaddr & 0xFFFFFFFFull);               // global_addr[31:0]
  g0.w = (unsigned)((gaddr >> 32) & 0x01FFFFFFull)        // global_addr[56:32]
       | (2u << 30);                                      // type=2 ("image")
  g1[0] = (int)(2u << 16);                                // wg_mask=0, data_size=4B
  g1[1] = (int)((dim0 & 0xFFFFu) << 16);                  // tensor_dim0 lo16
  g1[2] = (int)((dim0 >> 16) | (((unsigned)BLK) << 16));  // dim0 hi16 | tensor_dim1=128
  g1[3] = (int)(tile0 << 16);                             // dim1 hi16=0 | tile_dim0
  g1[4] = (int)BLK;                                       // tile_dim1=128, tile_dim2=0
  g1[5] = (int)stride0;                                   // tensor_dim0_stride lo32
  g1[6] = 0;
  g1[7] = 0;
}

__device__ __forceinline__ void tdm_load(unsigned lds_byte, unsigned long long gaddr,
                                         unsigned dim0, unsigned stride0, unsigned tile0) {
  v4u g0; v8i g1;
  tdm_desc(g0, g1, lds_byte, gaddr, dim0, stride0, tile0);
  v4i z4 = {0, 0, 0, 0};
#if defined(__clang_major__) && __clang_major__ >= 23
  v8i z8 = {0, 0, 0, 0, 0, 0, 0, 0};
  __builtin_amdgcn_tensor_load_to_lds(g0, g1, z4, z4, z8, 0);
#else
  __builtin_amdgcn_tensor_load_to_lds(g0, g1, z4, z4, 0);
#endif
}

__device__ __forceinline__ void tdm_store(unsigned lds_byte, unsigned long long gaddr,
                                          unsigned dim0, unsigned stride0, unsigned tile0) {
  v4u g0; v8i g1;
  tdm_desc(g0, g1, lds_byte, gaddr, dim0, stride0, tile0);
  v4i z4 = {0, 0, 0, 0};
#if defined(__clang_major__) && __clang_major__ >= 23
  v8i z8 = {0, 0, 0, 0, 0, 0, 0, 0};
  __builtin_amdgcn_tensor_store_from_lds(g0, g1, z4, z4, z8, 0);
#else
  __builtin_amdgcn_tensor_store_from_lds(g0, g1, z4, z4, 0);
#endif
}

// ---------------- small helpers ----------------
__device__ __forceinline__ void st4(float* p, float a, float b, float c, float d) {
  *reinterpret_cast<float4*>(p) = make_float4(a, b, c, d);
}

// Solve S u = b given chol factors (lower L; i** = reciprocals of diagonal).
__device__ __forceinline__ void chol_solve(
    float i00, float i11, float i22, float l10, float l20, float l21,
    float b0, float b1, float b2, float& u0, float& u1, float& u2) {
  float y0 = b0 * i00;
  float y1 = (b1 - l10 * y0) * i11;
  float y2 = (b2 - l20 * y0 - l21 * y1) * i22;
  u2 = y2 * i22;
  u1 = (y1 - l21 * u2) * i11;
  u0 = (y0 - l10 * u1 - l20 * u2) * i00;
}

// One EKF step (predict order-1 + Joseph update), H = I, R = r2*I, Q = q2*I.
// Writes post-update x into xb[0..2] (registers) and post-update P into pl[0..8] (LDS).
__device__ __forceinline__ void ekf_step(
    float z0, float z1, float z2,
    float& x1, float& x2, float& x3,
    float& p00, float& p01, float& p02, float& p11, float& p12, float& p22,
    float& nis_acc, float* xb, float* pl) {
  const float DTc = 0.02f, SG = 10.0f, RH = 28.0f, BE = 8.0f / 3.0f;
  const float R2c = 0.01f, Q2c = 0.001f, JIT = 1e-6f;

  // ---- predict ----
  float f00 = 1.0f - DTc * SG, f01 = DTc * SG,  f02 = 0.0f;
  float f10 = DTc * (RH - x3), f11 = 1.0f - DTc, f12 = -DTc * x1;
  float f20 = DTc * x2,        f21 = DTc * x1,  f22 = 1.0f - DTc * BE;
  float nx1 = x1 + DTc * (SG * (x2 - x1));
  float nx2 = x2 + DTc * (x1 * (RH - x3) - x2);
  float nx3 = x3 + DTc * (x1 * x2 - BE * x3);
  // A = F * P   (P symmetric)
  float a00 = f00 * p00 + f01 * p01 + f02 * p02;
  float a01 = f00 * p01 + f01 * p11 + f02 * p12;
  float a02 = f00 * p02 + f01 * p12 + f02 * p22;
  float a10 = f10 * p00 + f11 * p01 + f12 * p02;
  float a11 = f10 * p01 + f11 * p11 + f12 * p12;
  float a12 = f10 * p02 + f11 * p12 + f12 * p22;
  float a20 = f20 * p00 + f21 * p01 + f22 * p02;
  float a21 = f20 * p01 + f21 * p11 + f22 * p12;
  float a22 = f20 * p02 + f21 * p12 + f22 * p22;
  // Qm = A * Fᵀ ; P = 0.5(Qm+Qmᵀ) + q2*I
  float q00 = a00 * f00 + a01 * f01 + a02 * f02;
  float q01 = a00 * f10 + a01 * f11 + a02 * f12;
  float q02 = a00 * f20 + a01 * f21 + a02 * f22;
  float q10 = a10 * f00 + a11 * f01 + a12 * f02;
  float q11 = a10 * f10 + a11 * f11 + a12 * f12;
  float q12 = a10 * f20 + a11 * f21 + a12 * f22;
  float q20 = a20 * f00 + a21 * f01 + a22 * f02;
  float q21 = a20 * f10 + a21 * f11 + a22 * f12;
  float q22 = a20 * f20 + a21 * f21 + a22 * f22;
  p00 = q00 + Q2c;  p11 = q11 + Q2c;  p22 = q22 + Q2c;
  p01 = 0.5f * (q01 + q10);
  p02 = 0.5f * (q02 + q20);
  p12 = 0.5f * (q12 + q21);

  // ---- update (Joseph form) ----
  float iv0 = z0 - nx1, iv1 = z1 - nx2, iv2 = z2 - nx3;
  float s00 = p00 + (R2c + JIT), s11 = p11 + (R2c + JIT), s22 = p22 + (R2c + JIT);
  float s01 = p01, s02 = p02, s12 = p12;
  float l00 = sqrtf(s00);               float i00 = 1.0f / l00;
  float l10 = s01 * i00, l20 = s02 * i00;
  float l11 = sqrtf(s11 - l10 * l10);   float i11 = 1.0f / l11;
  float l21 = (s12 - l20 * l10) * i11;
  float l22 = sqrtf(s22 - l20 * l20 - l21 * l21);  float i22 = 1.0f / l22;
  // K = P S^-1 : row j of K solves S u = P[:,j]
  float k00, k01, k02, k10, k11, k12, k20, k21, k22;
  chol_solve(i00, i11, i22, l10, l20, l21, p00, p01, p02, k00, k01, k02);
  chol_solve(i00, i11, i22, l10, l20, l21, p01, p11, p12, k10, k11, k12);
  chol_solve(i00, i11, i22, l10, l20, l21, p02, p12, p22, k20, k21, k22);
  nx1 += k00 * iv0 + k01 * iv1 + k02 * iv2;
  nx2 += k10 * iv0 + k11 * iv1 + k12 * iv2;
  nx3 += k20 * iv0 + k21 * iv1 + k22 * iv2;
  // M = I - K
  float m00 = 1.0f - k00, m01 = -k01,        m02 = -k02;
  float m10 = -k10,       m11 = 1.0f - k11,  m12 = -k12;
  float m20 = -k20,       m21 = -k21,        m22 = 1.0f - k22;
  // B = M * P
  float b00 = m00 * p00 + m01 * p01 + m02 * p02;
  float b01 = m00 * p01 + m01 * p11 + m02 * p12;
  float b02 = m00 * p02 + m01 * p12 + m02 * p22;
  float b10 = m10 * p00 + m11 * p01 + m12 * p02;
  float b11 = m10 * p01 + m11 * p11 + m12 * p12;
  float b12 = m10 * p02 + m11 * p12 + m12 * p22;
  float b20 = m20 * p00 + m21 * p01 + m22 * p02;
  float b21 = m20 * p01 + m21 * p11 + m22 * p12;
  float b22 = m20 * p02 + m21 * p12 + m22 * p22;
  // C = B * Mᵀ
  float c00 = b00 * m00 + b01 * m01 + b02 * m02;
  float c01 = b00 * m10 + b01 * m11 + b02 * m12;
  float c02 = b00 * m20 + b01 * m21 + b02 * m22;
  float c10 = b10 * m00 + b11 * m01 + b12 * m02;
  float c11 = b10 * m10 + b11 * m11 + b12 * m12;
  float c12 = b10 * m20 + b11 * m21 + b12 * m22;
  float c20 = b20 * m00 + b21 * m01 + b22 * m02;
  float c21 = b20 * m10 + b21 * m11 + b22 * m12;
  float c22 = b20 * m20 + b21 * m21 + b22 * m22;
  // K R Kᵀ = r2 * K Kᵀ
  float w00 = k00 * k00 + k01 * k01 + k02 * k02;
  float w01 = k00 * k10 + k01 * k11 + k02 * k12;
  float w02 = k00 * k20 + k01 * k21 + k02 * k22;
  float w11 = k10 * k10 + k11 * k11 + k12 * k12;
  float w12 = k10 * k20 + k11 * k21 + k12 * k22;
  float w22 = k20 * k20 + k21 * k21 + k22 * k22;
  p00 = c00 + R2c * w00;
  p11 = c11 + R2c * w11;
  p22 = c22 + R2c * w22;
  p01 = 0.5f * (c01 + c10) + R2c * w01;
  p02 = 0.5f * (c02 + c20) + R2c * w02;
  p12 = 0.5f * (c12 + c21) + R2c * w12;
  // NIS
  float u0, u1, u2;
  chol_solve(i00, i11, i22, l10, l20, l21, iv0, iv1, iv2, u0, u1, u2);
  nis_acc += iv0 * u0 + iv1 * u1 + iv2 * u2;
  // commit
  x1 = nx1; x2 = nx2; x3 = nx3;
  xb[0] = nx1; xb[1] = nx2; xb[2] = nx3;
  pl[0] = p00; pl[1] = p01; pl[2] = p02;
  pl[3] = p01; pl[4] = p11; pl[5] = p12;
  pl[6] = p02; pl[7] = p12; pl[8] = p22;
}

// ---------------- main kernel: one lane per sequence ----------------
__global__ __launch_bounds__(BLK, 1) void ekf_lorenz_kernel(
    const float* __restrict__ Y, float* __restrict__ Xout,
    float* __restrict__ Pout, float* __restrict__ nis_part) {
  __shared__ float smem[SMEM_FLOATS];   // 2x Y staging (24 KB) + 2x P staging (72 KB)
  const int tid = threadIdx.x;
  const int n0  = blockIdx.x * BLK;
  const int n   = n0 + tid;

  // EKF state (reference initialize(): x0 = ones, P0 = 1e-5 I)
  float x1 = 1.0f, x2 = 1.0f, x3 = 1.0f;
  float p00 = 1e-5f, p11 = 1e-5f, p22 = 1e-5f, p01 = 0.0f, p02 = 0.0f, p12 = 0.0f;
  float nis_acc = 0.0f;

  float* Xrow = Xout + (size_t)n * (T_STEPS * 3);
  const unsigned lds_base = (unsigned)(uintptr_t)(const void*)smem;  // flat low32 == LDS offset
  const unsigned long long ybase =
      (unsigned long long)(uintptr_t)(Y + (size_t)n0 * (T_STEPS * 3));
  const unsigned long long pbase =
      (unsigned long long)(uintptr_t)(Pout + (size_t)n0 * (T_STEPS * 9));

  // prime the pipeline: Y chunk 0 -> Y buffer 0 (one TDM issue per block, wave 0)
  if (tid < 32) {
    tdm_load(lds_base, ybase, T_STEPS * 3, T_STEPS * 3, ROWF);
  }

  for (int c = 0; c < NCHUNK; ++c) {
    const int buf = c & 1;
    if (tid < 32) {
      // Tensor ops from one wave complete in order:
      //   tensorcnt<=1 => load(c) landed AND P-store(c-2) drained (buffer reusable).
      if (c == 0) __builtin_amdgcn_s_wait_tensorcnt(0);
      else        __builtin_amdgcn_s_wait_tensorcnt(1);
    }
    __syncthreads();                          // visibility + everyone done with Y buf^1
    if ((c + 1 < NCHUNK) && (tid < 32)) {     // prefetch Y chunk c+1 while computing c
      tdm_load(lds_base + (unsigned)((buf ^ 1) * YBUF * 4),
               ybase + (unsigned long long)((c + 1) * ROWF) * 4ull,
               T_STEPS * 3, T_STEPS * 3, ROWF);
    }
    const float* zp   = smem + buf * YBUF + tid * ROWF;
    float*       prow = smem + PSTG_OFF + buf * PBUF + tid * PROWF;

    for (int grp = 0; grp < 2; ++grp) {       // 2 groups of 4 steps per chunk
      float xb[12];
#pragma unroll
      for (int s4 = 0; s4 < 4; ++s4) {
        const int s = grp * 4 + s4;
        float z0 = zp[s * 3 + 0], z1 = zp[s * 3 + 1], z2 = zp[s * 3 + 2];
        ekf_step(z0, z1, z2, x1, x2, x3, p00, p01, p02, p11, p12, p22,
                 nis_acc, &xb[s4 * 3], prow + s * 9);
      }
      const int t0 = c * CHUNK + grp * 4;     // multiple of 4 -> float4 aligned
      float* xp = Xrow + t0 * 3;
#pragma unroll
      for (int k = 0; k < 3; ++k)
        st4(xp + 4 * k, xb[4 * k], xb[4 * k + 1], xb[4 * k + 2], xb[4 * k + 3]);
    }

    __syncthreads();                          // all lanes' P rows for chunk c are in LDS
    if (tid < 32) {                           // async scatter: LDS -> P_hat[n0:n0+128, c*8:(c+1)*8, :, :]
      tdm_store(lds_base + (unsigned)((PSTG_OFF + buf * PBUF) * 4),
                pbase + (unsigned long long)(c * PROWF) * 4ull,
                T_STEPS * 9, T_STEPS * 9, PROWF);
    }
  }

  // deterministic per-block NIS reduction (reuse Y staging LDS; disjoint from P staging
  // that the final in-flight tensor_store is still reading; S_ENDPGM waits all counters)
  __syncthreads();
  float* red = smem;
  red[tid] = nis_acc;
  __syncthreads();
#pragma unroll
  for (int off = BLK / 2; off > 0; off >>= 1) {
    if (tid < off) red[tid] += red[tid + off];
    __syncthreads();
  }
  if (tid == 0) nis_part[blockIdx.x] = red[0];
}

// ---------------- finalize: mean of NIS over N*T ----------------
__global__ void nis_finalize(const float* __restrict__ part, float* __restrict__ out) {
  if (threadIdx.x == 0 && blockIdx.x == 0) {
    float s = 0.0f;
    for (int i = 0; i < GRID_B; ++i) s += part[i];
    out[(size_t)N_SEQ * T_STEPS * 12] = s / (float)((size_t)N_SEQ * T_STEPS);
  }
}

extern "C" void kernel_launch(void* const* d_in, const int* in_sizes, int n_in,
                              void* d_out, int out_size, void* d_ws, size_t ws_size,
                              hipStream_t stream) {
  (void)in_sizes; (void)n_in; (void)out_size; (void)ws_size;
  const float* Y = (const float*)d_in[0];   // [4096, 1000, 3] f32 (Q,R,H are fixed constants)
  float* out  = (float*)d_out;              // X_hat | P_hat | nis_mean, concatenated flat
  float* Xout = out;
  float* Pout = out + (size_t)N_SEQ * T_STEPS * 3;
  float* part = (float*)d_ws;               // 32 block partials
  ekf_lorenz_kernel<<<GRID_B, BLK, 0, stream>>>(Y, Xout, Pout, part);
  nis_finalize<<<1, 1, 0, stream>>>(part, out);
}